// Logic_Learning_Model_76295799046425
// MI455X (gfx1250) — compile-verified
//
#include <hip/hip_runtime.h>
#include <hip/hip_bf16.h>

typedef __attribute__((ext_vector_type(2))) float v2f;
typedef __attribute__((ext_vector_type(8))) float v8f;

#define TOL_C   0.5f
#define RES_C   0.03f
#define NB      128
#define NN      64
#define NH      128
#define NG      4000
#define GRID_TILES 250          /* 4000 / 16 */
#define EV_TILES   8            /* ceil(127 / 16) */
#define TOT_TILES  (GRID_TILES + EV_TILES)

// t value for slot `idx` (0..15) of tile `lt`.
// Grid tiles: t = (16*lt+idx)*RES.  Event tiles: t = head_times[1 + 16*lt + idx],
// valid only while the event index stays below H-1 = 127.
__device__ __forceinline__ float tval_of(bool isEvent, int lt, int idx,
                                         const float* hts, bool* valid) {
  if (!isEvent) { *valid = true; return (float)(lt * 16 + idx) * RES_C; }
  int e = lt * 16 + idx;                 // 0..127, valid events are 0..126
  *valid = (e < NH - 1);
  int h = e + 1; if (h > NH - 1) h = NH - 1;
  return hts[h];
}

__global__ void __launch_bounds__(256)
ll_intensity_kernel(const float* __restrict__ times0, const int* __restrict__ states0,
                    const float* __restrict__ times1, const int* __restrict__ states1,
                    const float* __restrict__ head_times, const int* __restrict__ head_states,
                    const float* __restrict__ base, const float* __restrict__ weights,
                    float* __restrict__ ws) {
  __shared__ float t0s[NN]; __shared__ int s0s[NN];
  __shared__ float t1s[NN]; __shared__ int s1s[NN];
  __shared__ float hts[NH]; __shared__ int hss[NH];
  __shared__ float red[256];

  const int b   = blockIdx.x;      // one block per batch
  const int tid = threadIdx.x;

  if (tid < NN) {
    t0s[tid] = times0[b*NN + tid]; s0s[tid] = states0[b*NN + tid];
    t1s[tid] = times1[b*NN + tid]; s1s[tid] = states1[b*NN + tid];
  }
  if (tid < NH) { hts[tid] = head_times[b*NH + tid]; hss[tid] = head_states[b*NH + tid]; }
  __syncthreads();

  // softmax of the two weights (one-time: keep accurate expf)
  const float ew0 = expf(weights[0]), ew1 = expf(weights[1]);
  const float w0  = ew0 / (ew0 + ew1);
  const float w1  = ew1 / (ew0 + ew1);
  const float bs  = base[0];

  const int wave = tid >> 5;       // 8 waves per block
  const int lane = tid & 31;
  const int half = lane >> 4;      // 0: lanes 0-15, 1: lanes 16-31
  const int lc   = lane & 15;

  float waveAcc = 0.f;

  for (int tile = wave; tile < TOT_TILES; tile += 8) {
    const bool isEvent = (tile >= GRID_TILES);
    const int  lt      = isEvent ? (tile - GRID_TILES) : tile;

    // --- t for the A-matrix row this lane feeds (A 16x4: lane&15 = row M) ---
    bool dum;
    const float tA = tval_of(isEvent, lt, lc, hts, &dum);

    // --- per-lane times0 for the 4 i-tiles (B/C column = lane&15 + 16*q) ---
    float t0i[4]; int s0i[4];
#pragma unroll
    for (int q = 0; q < 4; ++q) { t0i[q] = t0s[16*q + lc]; s0i[q] = s0s[16*q + lc]; }

    // --- t for the C-layout rows this lane owns (row M = r + 8*half) ---
    float tRow[8]; bool vRow[8];
#pragma unroll
    for (int r = 0; r < 8; ++r) tRow[r] = tval_of(isEvent, lt, r + 8*half, hts, &vRow[r]);

    // --- V[t,i] = sum_j Bm[t,j] * M[i,j]  via fp32 WMMA 16x16x4, K chunks of 4 ---
    v8f acc[4];
#pragma unroll
    for (int q = 0; q < 4; ++q) acc[q] = (v8f){0.f,0.f,0.f,0.f,0.f,0.f,0.f,0.f};

#pragma unroll
    for (int k = 0; k < 16; ++k) {
      const int   j0  = 4*k + 2*half;          // lane's K pair: {j0, j0+1}
      const float T1a = t1s[j0],   T1b = t1s[j0+1];
      const int   S1a = s1s[j0],   S1b = s1s[j0+1];
      const float da  = tA - T1a,  db  = tA - T1b;
      const float ea  = __expf(-fmaxf(da, 0.f));   // single v_exp_f32
      const float eb  = __expf(-fmaxf(db, 0.f));
      v2f a;                                    // A = Bm tile (16 t x 4 j)
      a.x = ((da > TOL_C) && (S1a == 1)) ? ea : 0.f;
      a.y = ((db > TOL_C) && (S1b == 1)) ? eb : 0.f;
#pragma unroll
      for (int q = 0; q < 4; ++q) {             // B = M^T slice (4 j x 16 i)
        v2f bm;
        bm.x = (t0i[q] - T1a < -TOL_C) ? 1.f : 0.f;
        bm.y = (t0i[q] - T1b < -TOL_C) ? 1.f : 0.f;
        acc[q] = __builtin_amdgcn_wmma_f32_16x16x4_f32(
            false, a, false, bm, (short)0, acc[q], false, false);
      }
    }

    // --- feat0 partial: V[t,i]*Adec[t,i]; feat1 partial: masked decays over i ---
    float p0[8], p1[8];
#pragma unroll
    for (int r = 0; r < 8; ++r) { p0[r] = 0.f; p1[r] = 0.f; }
#pragma unroll
    for (int q = 0; q < 4; ++q) {
      const float T0 = t0i[q]; const int S0 = s0i[q];
#pragma unroll
      for (int r = 0; r < 8; ++r) {
        const float d = tRow[r] - T0;
        const float e = __expf(-fmaxf(d, 0.f));  // single v_exp_f32
        p0[r] += ((d >= 0.f)   && (S0 == 1)) ? acc[q][r] * e : 0.f;  // A mask
        p1[r] += ((d >  TOL_C) && (S0 == 0)) ? e : 0.f;              // feat1 mask
      }
    }

    // --- reduce over i within each 16-lane half, then score each t ---
#pragma unroll
    for (int r = 0; r < 8; ++r) {
      float f0 = p0[r], f1 = p1[r];
#pragma unroll
      for (int m = 1; m <= 8; m <<= 1) {
        f0 += __shfl_xor(f0, m, 32);
        f1 += __shfl_xor(f1, m, 32);
      }
      if (lc == 0 && vRow[r]) {
        const float t = tRow[r];
        // branchless lower-bound: cnt = #{h : head_times[h] < t}
        int cnt = 0;
#pragma unroll
        for (int s = 128; s > 0; s >>= 1) {
          int probe = cnt + s;
          if (probe <= NH && hts[probe - 1] < t) cnt = probe;
        }
        const int   st   = hss[(cnt + NH - 1) & (NH - 1)];  // (cnt-1) mod 128, Python-style
        const float eff0 = (st == 0) ? 1.f : -1.f;
        const float ll   = bs + eff0 * (w0 * f0 - w1 * f1); // eff1 = -eff0
        waveAcc += isEvent ? ll : (-RES_C * __expf(ll));
      }
    }
  }

  // deterministic block reduction
  red[tid] = waveAcc;
  __syncthreads();
  for (int s = 128; s > 0; s >>= 1) {
    if (tid < s) red[tid] += red[tid + s];
    __syncthreads();
  }
  if (tid == 0) ws[b] = red[0];
}

__global__ void __launch_bounds__(128)
ll_finalize_kernel(const float* __restrict__ ws, float* __restrict__ out) {
  __shared__ float red[NB];
  const int tid = threadIdx.x;
  red[tid] = ws[tid];
  __syncthreads();
  for (int s = 64; s > 0; s >>= 1) {
    if (tid < s) red[tid] += red[tid + s];
    __syncthreads();
  }
  if (tid == 0) out[0] = red[0];
}

extern "C" void kernel_launch(void* const* d_in, const int* in_sizes, int n_in,
                              void* d_out, int out_size, void* d_ws, size_t ws_size,
                              hipStream_t stream) {
  (void)in_sizes; (void)n_in; (void)out_size; (void)ws_size;
  const float* times0      = (const float*)d_in[0];
  const int*   states0     = (const int*)  d_in[1];
  const float* times1      = (const float*)d_in[2];
  const int*   states1     = (const int*)  d_in[3];
  const float* head_times  = (const float*)d_in[4];
  const int*   head_states = (const int*)  d_in[5];
  const float* base        = (const float*)d_in[6];
  const float* weights     = (const float*)d_in[7];
  float* out = (float*)d_out;
  float* wsf = (float*)d_ws;   // 128 floats of scratch

  ll_intensity_kernel<<<dim3(NB), dim3(256), 0, stream>>>(
      times0, states0, times1, states1, head_times, head_states, base, weights, wsf);
  ll_finalize_kernel<<<dim3(1), dim3(128), 0, stream>>>(wsf, out);
}